// STGNN_64218351010250
// MI455X (gfx1250) — compile-verified
//
#include <hip/hip_runtime.h>
#include <stdint.h>

#define GN_N 50000
#define GN_E 800000
#define GN_D 96
#define GN_K 3
#define EPB 8   // edges per block (one wave32 per edge, 256-thread block)

// ---- degree: deg[col[e]] += 1 ------------------------------------------------
__global__ void deg_kernel(const int* __restrict__ col, float* __restrict__ deg, int E) {
    int e = blockIdx.x * blockDim.x + threadIdx.x;
    if (e < E) atomicAdd(&deg[col[e]], 1.0f);
}

// ---- deg_inv = deg > 0 ? deg^-0.5 : 0 ---------------------------------------
__global__ void deginv_kernel(float* __restrict__ deg, int N) {
    int n = blockIdx.x * blockDim.x + threadIdx.x;
    if (n < N) {
        float d = deg[n];
        deg[n] = (d > 0.0f) ? (1.0f / sqrtf(d)) : 0.0f;
    }
}

// ---- norm[e] = deg_inv[row[e]] * deg_inv[col[e]] ----------------------------
__global__ void norm_kernel(const int* __restrict__ row, const int* __restrict__ col,
                            const float* __restrict__ dinv, float* __restrict__ nrm, int E) {
    int e = blockIdx.x * blockDim.x + threadIdx.x;
    if (e < E) nrm[e] = dinv[row[e]] * dinv[col[e]];
}

// ---- hidden = hopwise[0] * x (vectorized) -----------------------------------
__global__ void init_kernel(const float4* __restrict__ x, const float* __restrict__ hopwise,
                            float4* __restrict__ out, int n4) {
    int i = blockIdx.x * blockDim.x + threadIdx.x;
    if (i < n4) {
        float h = hopwise[0];
        float4 v = x[i];
        float4 r; r.x = h * v.x; r.y = h * v.y; r.z = h * v.z; r.w = h * v.w;
        out[i] = r;
    }
}

// ---- hidden += hopwise[h] * buf (vectorized) --------------------------------
__global__ void axpy_kernel(const float4* __restrict__ b, const float* __restrict__ hopwise,
                            int h, float4* __restrict__ out, int n4) {
    int i = blockIdx.x * blockDim.x + threadIdx.x;
    if (i < n4) {
        float hw = hopwise[h];
        float4 v = b[i];
        float4 o = out[i];
        o.x += hw * v.x; o.y += hw * v.y; o.z += hw * v.z; o.w += hw * v.w;
        out[i] = o;
    }
}

// ---- one propagation hop: xdst[col] += norm * (xsrc[row] + ef) --------------
// One wave32 per edge; lanes 0..23 each own one float4 of the 96-wide row.
// edge_feature for the block's 8 consecutive edges is a contiguous 3 KB run:
// staged into LDS with CDNA5 async load-to-LDS (ASYNCcnt), marked NT so the
// 307 MB stream does not evict the L2-resident gather/scatter arrays.
__global__ __launch_bounds__(256) void prop_kernel(
        const float* __restrict__ xsrc, const float* __restrict__ ef,
        const int* __restrict__ row, const int* __restrict__ col,
        const float* __restrict__ nrm, float* __restrict__ xdst, int E) {
    __shared__ float efs[EPB * GN_D];   // 3072 B

    const int t = threadIdx.x;
    const long long base = (long long)blockIdx.x * EPB;
    const int nEdges = (int)(((long long)E - base) < EPB ? ((long long)E - base) : EPB);
    const int nVec = nEdges * (GN_D / 4);      // float4 chunks to stage (<=192)

    // Stage 1: issue async copies global -> LDS (b128 per lane, non-temporal).
    if (t < nVec) {
        const float* gsrc = ef + base * GN_D + (long long)t * 4;
        unsigned lds_off = (unsigned)(uintptr_t)(&efs[t * 4]);
        asm volatile("global_load_async_to_lds_b128 %0, %1, off th:TH_LOAD_NT"
                     :: "v"(lds_off), "v"(gsrc) : "memory");
    }

    // Stage 2: wave-uniform edge metadata via scalar path (readfirstlane ->
    // s_load on the KMcnt/constant-cache pipe), overlapping the async stream.
    const int w = t >> 5;          // wave id within block = edge slot (uniform)
    const int lane = t & 31;
    const int ew = __builtin_amdgcn_readfirstlane(w);   // force SGPR
    const bool haveEdge = (ew < nEdges);
    const bool active = haveEdge && (lane < (GN_D / 4));

    int r = 0, c = 0;
    float nv = 0.0f;
    if (haveEdge) {
        long long e = base + ew;
        r  = row[e];
        c  = col[e];
        nv = nrm[e];
    }
    float4 xv = make_float4(0.f, 0.f, 0.f, 0.f);
    if (active) {
        xv = ((const float4*)xsrc)[(long long)r * (GN_D / 4) + lane];
    }

    // Stage 3: drain ASYNCcnt, make LDS visible block-wide.
    asm volatile("s_wait_asynccnt 0" ::: "memory");
    __syncthreads();

    // Stage 4: compute message and scatter-add (f32 atomics, L2-resident dst).
    if (active) {
        float4 efv = ((const float4*)efs)[ew * (GN_D / 4) + lane];
        float* dst = xdst + (long long)c * GN_D + lane * 4;
        atomicAdd(dst + 0, nv * (xv.x + efv.x));
        atomicAdd(dst + 1, nv * (xv.y + efv.y));
        atomicAdd(dst + 2, nv * (xv.z + efv.z));
        atomicAdd(dst + 3, nv * (xv.w + efv.w));
    }
}

extern "C" void kernel_launch(void* const* d_in, const int* in_sizes, int n_in,
                              void* d_out, int out_size, void* d_ws, size_t ws_size,
                              hipStream_t stream) {
    (void)in_sizes; (void)n_in; (void)out_size; (void)ws_size;

    const float* x       = (const float*)d_in[0];   // [N, D]
    const float* ef      = (const float*)d_in[1];   // [E, D]
    const float* hopwise = (const float*)d_in[2];   // [K+1]
    const int*   ei      = (const int*)d_in[3];     // [2, E]
    const int* row = ei;
    const int* col = ei + GN_E;

    float* ws     = (float*)d_ws;
    float* deginv = ws;                              // N
    float* nrm    = deginv + GN_N;                   // E
    float* bufA   = nrm + GN_E;                      // N*D
    float* bufB   = bufA + (size_t)GN_N * GN_D;      // N*D
    float* hidden = (float*)d_out;                   // N*D

    const int n4 = GN_N * GN_D / 4;
    const size_t xbytes = (size_t)GN_N * GN_D * sizeof(float);

    // Normalization (recomputed every call; deterministic).
    hipMemsetAsync(deginv, 0, GN_N * sizeof(float), stream);
    deg_kernel   <<<(GN_E + 255) / 256, 256, 0, stream>>>(col, deginv, GN_E);
    deginv_kernel<<<(GN_N + 255) / 256, 256, 0, stream>>>(deginv, GN_N);
    norm_kernel  <<<(GN_E + 255) / 256, 256, 0, stream>>>(row, col, deginv, nrm, GN_E);

    // hidden = hopwise[0] * x
    init_kernel<<<(n4 + 255) / 256, 256, 0, stream>>>(
        (const float4*)x, hopwise, (float4*)hidden, n4);

    // K hops, ping-ponging workspace buffers; hop 0 gathers from original x.
    const float* src = x;
    float* dsts[GN_K] = { bufB, bufA, bufB };
    for (int hop = 0; hop < GN_K; ++hop) {
        float* dst = dsts[hop];
        hipMemsetAsync(dst, 0, xbytes, stream);
        prop_kernel<<<(GN_E + EPB - 1) / EPB, 256, 0, stream>>>(
            src, ef, row, col, nrm, dst, GN_E);
        axpy_kernel<<<(n4 + 255) / 256, 256, 0, stream>>>(
            (const float4*)dst, hopwise, hop + 1, (float4*)hidden, n4);
        src = dst;
    }
}